// SSIMLoss_90726889161515
// MI455X (gfx1250) — compile-verified
//
#include <hip/hip_runtime.h>

// SSIM loss, fused single-pass kernel using fp32 WMMA (banded-matrix blur).
// pred/target: [16,3,512,512] f32.  Output: scalar f32 = 1 - mean(ssim_map).

typedef __attribute__((ext_vector_type(2))) float v2f;
typedef __attribute__((ext_vector_type(8))) float v8f;

#define IMG_W   512
#define NBLK    32          // 512/16 blocks per image dimension
#define ROWPAD  50          // LDS rows per column: 3 ring blocks * 16 = 48, padded
#define SSIM_C1 1.0e-4f
#define SSIM_C2 9.0e-4f

__device__ __forceinline__ v8f wmma_f32(v2f a, v2f b, v8f c) {
    // D(16x16,f32) = A(16x4,f32) * B(4x16,f32) + C
    return __builtin_amdgcn_wmma_f32_16x16x4_f32(false, a, false, b, (short)0, c,
                                                 false, false);
}

__global__ __launch_bounds__(128) void ssim_main_kernel(
    const float* __restrict__ pred,
    const float* __restrict__ targ,
    float* __restrict__ partial)
{
    // HB ring: [quantity(5)][col within 64-wide tile][row ring 48 (pad 50)]
    __shared__ float hbuf[5 * 64 * ROWPAD];    // 64000 bytes
    __shared__ float wsh[11];

    const int tid  = threadIdx.x;
    const int wave = tid >> 5;     // 0..3, owns one 16-wide output column block
    const int lane = tid & 31;
    const int hh   = lane >> 4;    // lane half (K-split for A/B operands)
    const int mm   = lane & 15;    // row index (A/C/D) or column index (B)

    // ---- Gaussian taps (sigma=1.5, 11 wide), normalized 1D window ----
    if (tid < 11) {
        float x = (float)tid - 5.0f;
        wsh[tid] = __expf(-(x * x) / 4.5f);
    }
    __syncthreads();
    float wsum = 0.0f;
#pragma unroll
    for (int i = 0; i < 11; ++i) wsum += wsh[i];
    const float rws = 1.0f / wsum;

    // ---- Band-matrix chunks: only 8 of 12 (blockOffset, K-step) chunks
    //      intersect the 11-wide band. band[p] serves as the constant
    //      B operand (hblur) and A operand (vblur) - Gaussian is symmetric.
    const int BTAB[8] = {-1, -1, 0, 0, 0, 0, 1, 1};
    const int STAB[8] = { 2,  3, 0, 1, 2, 3, 0, 1};
    v2f band[8];
#pragma unroll
    for (int p = 0; p < 8; ++p) {
        int kk0 = STAB[p] * 4 + 2 * hh;
        int d0  = BTAB[p] * 16 + kk0 - mm;
        int d1  = d0 + 1;
        band[p].x = (d0 >= -5 && d0 <= 5) ? wsh[d0 + 5] * rws : 0.0f;
        band[p].y = (d1 >= -5 && d1 <= 5) ? wsh[d1 + 5] * rws : 0.0f;
    }

    const int    CB      = blockIdx.x * 4 + wave;       // global col block 0..31
    const int    RB0     = blockIdx.y * 4;              // first output row block
    const size_t imgbase = (size_t)blockIdx.z * (IMG_W * IMG_W);
    const int    colL    = wave * 16 + mm;              // column within 64-wide tile

    float lsum = 0.0f;

    // Pipeline over horizontal-blur row blocks RB0-1 .. RB0+4.
    for (int it = 0; it < 6; ++it) {
        const int  hb_rb   = RB0 - 1 + it;
        const bool hbValid = (hb_rb >= 0) && (hb_rb < NBLK);

        v8f acc[5];
#pragma unroll
        for (int q = 0; q < 5; ++q) acc[q] = (v8f){};

        // ---- Horizontal blur: HB = X * G (A = data, B = band chunk) ----
        if (hbValid) {
            const int row = hb_rb * 16 + mm;
#pragma unroll
            for (int p = 0; p < 8; ++p) {
                const int kbg = CB + BTAB[p];           // input col block (uniform)
                if (kbg < 0 || kbg >= NBLK) continue;   // zero padding
                const int    col = kbg * 16 + STAB[p] * 4 + 2 * hh;
                const size_t idx = imgbase + (size_t)row * IMG_W + col;
                v2f pv = *reinterpret_cast<const v2f*>(pred + idx);
                v2f tv = *reinterpret_cast<const v2f*>(targ + idx);
                pv = (pv + 1.0f) * 0.5f;
                tv = (tv + 1.0f) * 0.5f;
                v2f a0 = pv, a1 = tv, a2 = pv * pv, a3 = tv * tv, a4 = pv * tv;
                acc[0] = wmma_f32(a0, band[p], acc[0]);
                acc[1] = wmma_f32(a1, band[p], acc[1]);
                acc[2] = wmma_f32(a2, band[p], acc[2]);
                acc[3] = wmma_f32(a3, band[p], acc[3]);
                acc[4] = wmma_f32(a4, band[p], acc[4]);
            }
        }

        __syncthreads();   // previous vblur done reading the ring slot we overwrite

        // Store HB transposed into the ring: [q][col][rowRing]
        if (hbValid) {
            const int rowm = (hb_rb % 3) * 16;
#pragma unroll
            for (int q = 0; q < 5; ++q) {
#pragma unroll
                for (int i = 0; i < 8; ++i) {
                    // D layout: vgpr i holds output row i (+8 for upper lane half)
                    hbuf[(q * 64 + colL) * ROWPAD + rowm + i + 8 * hh] = acc[q][i];
                }
            }
        }

        __syncthreads();

        // ---- Vertical blur + SSIM for out_rb = hb_rb - 1 ----
        const int out_rb = hb_rb - 1;
        if (out_rb >= RB0 && out_rb <= RB0 + 3) {
            v8f vacc[5];
#pragma unroll
            for (int q = 0; q < 5; ++q) vacc[q] = (v8f){};

#pragma unroll
            for (int p = 0; p < 8; ++p) {
                const int hbb = out_rb + BTAB[p];       // HB row block (uniform)
                if (hbb < 0 || hbb >= NBLK) continue;   // zero padding
                const int rbase = (hbb % 3) * 16 + STAB[p] * 4 + 2 * hh;
#pragma unroll
                for (int q = 0; q < 5; ++q) {
                    v2f bhb = *reinterpret_cast<const v2f*>(
                        &hbuf[(q * 64 + colL) * ROWPAD + rbase]);
                    vacc[q] = wmma_f32(band[p], bhb, vacc[q]);
                }
            }

            // SSIM map on this lane's 8 accumulator rows
#pragma unroll
            for (int i = 0; i < 8; ++i) {
                float mp  = vacc[0][i];
                float mt  = vacc[1][i];
                float spp = vacc[2][i] - mp * mp;
                float stt = vacc[3][i] - mt * mt;
                float spt = vacc[4][i] - mp * mt;
                float num = (2.0f * mp * mt + SSIM_C1) * (2.0f * spt + SSIM_C2);
                float den = (mp * mp + mt * mt + SSIM_C1) * (spp + stt + SSIM_C2);
                lsum += num / den;
            }
        }
    }

    // ---- Reduce: wave shuffle, then per-workgroup partial to workspace ----
#pragma unroll
    for (int off = 16; off > 0; off >>= 1)
        lsum += __shfl_down(lsum, off, 32);

    __syncthreads();                   // hbuf no longer needed; reuse for 4 floats
    if (lane == 0) hbuf[wave] = lsum;
    __syncthreads();
    if (tid == 0) {
        float s = hbuf[0] + hbuf[1] + hbuf[2] + hbuf[3];
        partial[(size_t)blockIdx.z * 64 + blockIdx.y * 8 + blockIdx.x] = s;
    }
}

__global__ __launch_bounds__(256) void ssim_reduce_kernel(
    const float* __restrict__ partial, float* __restrict__ out, int n)
{
    __shared__ double red[256];
    double s = 0.0;
    for (int i = threadIdx.x; i < n; i += 256) s += (double)partial[i];
    red[threadIdx.x] = s;
    __syncthreads();
    for (int off = 128; off > 0; off >>= 1) {
        if (threadIdx.x < off) red[threadIdx.x] += red[threadIdx.x + off];
        __syncthreads();
    }
    if (threadIdx.x == 0) {
        const double N = 12582912.0;   // 16*3*512*512
        out[0] = 1.0f - (float)(red[0] / N);
    }
}

extern "C" void kernel_launch(void* const* d_in, const int* in_sizes, int n_in,
                              void* d_out, int out_size, void* d_ws, size_t ws_size,
                              hipStream_t stream) {
    (void)in_sizes; (void)n_in; (void)out_size; (void)ws_size;
    const float* pred    = (const float*)d_in[0];
    const float* targ    = (const float*)d_in[1];
    float*       out     = (float*)d_out;
    float*       partial = (float*)d_ws;          // 3072 floats = 12 KB

    dim3 grid(8, 8, 48);   // 8x8 tiles of 64x64 per image, 48 images
    ssim_main_kernel<<<grid, 128, 0, stream>>>(pred, targ, partial);
    ssim_reduce_kernel<<<1, 256, 0, stream>>>(partial, out, 8 * 8 * 48);
}